// GCNet_83219286328194
// MI455X (gfx1250) — compile-verified
//
#include <hip/hip_runtime.h>

typedef __attribute__((ext_vector_type(2))) float v2f;
typedef __attribute__((ext_vector_type(8))) float v8f;

__device__ __forceinline__ void atomicAddF(float* p, float v) {
    (void)__hip_atomic_fetch_add(p, v, __ATOMIC_RELAXED, __HIP_MEMORY_SCOPE_AGENT);
}

// ---------------- normalization precompute ----------------

__global__ __launch_bounds__(256) void deg_init(float* __restrict__ deg, int n) {
    int i = blockIdx.x * blockDim.x + threadIdx.x;
    if (i < n) deg[i] = 2.0f;  // self-loop weight (improved=True)
}

__global__ __launch_bounds__(256) void deg_edges(const int* __restrict__ dst,
                                                 const float* __restrict__ w,
                                                 float* __restrict__ deg, int E) {
    int e = blockIdx.x * blockDim.x + threadIdx.x;
    if (e < E) atomicAddF(&deg[dst[e]], w[e]);
}

__global__ __launch_bounds__(256) void make_dinv(float* __restrict__ deg, int n) {
    int i = blockIdx.x * blockDim.x + threadIdx.x;
    if (i < n) {
        float d = deg[i];
        deg[i] = d > 0.0f ? rsqrtf(d) : 0.0f;
    }
}

__global__ __launch_bounds__(256) void make_norm(const int* __restrict__ src,
                                                 const int* __restrict__ dst,
                                                 const float* __restrict__ w,
                                                 const float* __restrict__ dinv,
                                                 float* __restrict__ norm, int E) {
    int e = blockIdx.x * blockDim.x + threadIdx.x;
    if (e < E) norm[e] = dinv[src[e]] * w[e] * dinv[dst[e]];
}

// ---------------- WMMA fp32 GEMM: H = (relu?)X @ W ----------------
// One wave computes a 16-row stripe across all NOUT columns.
// W is staged TRANSPOSED in LDS with an XOR swizzle at float2 granularity:
//   physical(n,k) = n*KDIM + (((k&~1) ^ ((2n)&(KDIM-1)&~1)) | (k&1))
// -> each B fragment is one aligned ds_load_b64, lanes spread over all 64 banks.

template <int KDIM, int NOUT, bool RELU_IN>
__global__ __launch_bounds__(256) void gemm_wmma(const float* __restrict__ X,
                                                 const float* __restrict__ W,
                                                 float* __restrict__ H, int nrows) {
    extern __shared__ float WsT[];  // NOUT * KDIM floats (<= 64KB)
    for (int i = threadIdx.x; i < KDIM * NOUT; i += 256) {
        int k = i / NOUT;
        int n = i - k * NOUT;
        int xv = (2 * n) & (KDIM - 1) & ~1;
        WsT[n * KDIM + (((k & ~1) ^ xv) | (k & 1))] = W[i];
    }
    __syncthreads();

    constexpr int NT = NOUT / 16;  // column tiles
    const int wid = blockIdx.x * 8 + (threadIdx.x >> 5);
    const int rowBase = wid * 16;
    if (rowBase >= nrows) return;  // uniform per wave: EXEC stays all-1s

    const int lane = threadIdx.x & 31;
    const int m = lane & 15;       // A-row within tile / C-column within tile
    const int half = lane >> 4;

    // Per-column-tile swizzle constants and column base pointers
    const float* colp[NT];
    int xv[NT];
#pragma unroll
    for (int t = 0; t < NT; ++t) {
        int n = t * 16 + m;
        xv[t] = (2 * n) & (KDIM - 1) & ~1;
        colp[t] = WsT + n * KDIM;
    }

    v8f acc[NT] = {};

    const float* xrow = X + (size_t)(rowBase + m) * KDIM;
    for (int k0 = 0; k0 < KDIM; k0 += 4) {
        // A 16x4 fp32 layout: VGPR0 = K=(0|2), VGPR1 = K=(1|3) per half-wave
        float2 av = *(const float2*)(xrow + k0 + 2 * half);
        if (RELU_IN) {
            av.x = fmaxf(av.x, 0.0f);
            av.y = fmaxf(av.y, 0.0f);
        }
        v2f a;
        a.x = av.x;
        a.y = av.y;
        const int krow = k0 + 2 * half;  // even
#pragma unroll
        for (int t = 0; t < NT; ++t) {
            const float2 bv = *(const float2*)(colp[t] + (krow ^ xv[t]));
            v2f b;  // B 4x16: row striped across lanes within a VGPR
            b.x = bv.x;
            b.y = bv.y;
            acc[t] = __builtin_amdgcn_wmma_f32_16x16x4_f32(
                false, a, false, b, (short)0, acc[t], false, false);
        }
    }

#pragma unroll
    for (int t = 0; t < NT; ++t) {
#pragma unroll
        for (int v = 0; v < 8; ++v) {
            H[(size_t)(rowBase + v + 8 * half) * NOUT + t * 16 + m] = acc[t][v];
        }
    }
}

// ---------------- aggregation: Y = bias + selfnorm*H + scatter ----------------

template <int C>
__global__ __launch_bounds__(256) void agg_init(const float* __restrict__ H,
                                                const float* __restrict__ dinv,
                                                const float* __restrict__ bias,
                                                float* __restrict__ Y, int n) {
    size_t idx = (size_t)blockIdx.x * blockDim.x + threadIdx.x;
    if (idx >= (size_t)n * C) return;
    int i = (int)(idx / C);
    int c = (int)(idx % C);
    float di = dinv[i];
    // self-loop: norm = dinv*2.0*dinv
    Y[idx] = bias[c] + (2.0f * di * di) * H[idx];
}

// C == 128: one wave per edge; edge id forced wave-uniform so the index /
// norm reads become scalar (SMEM) loads, lanes do the float4 gather + atomics.
__global__ __launch_bounds__(256) void agg_edges128(const int* __restrict__ src,
                                                    const int* __restrict__ dst,
                                                    const float* __restrict__ norm,
                                                    const float* __restrict__ H,
                                                    float* __restrict__ Y, int E) {
    int e = blockIdx.x * 8 + (threadIdx.x >> 5);
    if (e >= E) return;  // uniform per wave
    e = __builtin_amdgcn_readfirstlane(e);
    const int lane = threadIdx.x & 31;
    const float nm = norm[e];
    const int s = src[e];
    const int d = dst[e];
    const float4 h = *(const float4*)(H + (size_t)s * 128 + lane * 4);
    float* y = Y + (size_t)d * 128 + lane * 4;
    atomicAddF(y + 0, nm * h.x);
    atomicAddF(y + 1, nm * h.y);
    atomicAddF(y + 2, nm * h.z);
    atomicAddF(y + 3, nm * h.w);
}

// Generic (used for C == 16): thread per (edge, float4 chunk)
template <int C>
__global__ __launch_bounds__(256) void agg_edges(const int* __restrict__ src,
                                                 const int* __restrict__ dst,
                                                 const float* __restrict__ norm,
                                                 const float* __restrict__ H,
                                                 float* __restrict__ Y, int E) {
    constexpr int CH4 = C / 4;
    size_t t = (size_t)blockIdx.x * blockDim.x + threadIdx.x;
    if (t >= (size_t)E * CH4) return;
    int e = (int)(t / CH4);
    int c4 = (int)(t % CH4) * 4;
    float nm = norm[e];
    int s = src[e], d = dst[e];
    const float4 h = *(const float4*)(H + (size_t)s * C + c4);
    float* y = Y + (size_t)d * C + c4;
    atomicAddF(y + 0, nm * h.x);
    atomicAddF(y + 1, nm * h.y);
    atomicAddF(y + 2, nm * h.z);
    atomicAddF(y + 3, nm * h.w);
}

// ---------------- launcher ----------------

extern "C" void kernel_launch(void* const* d_in, const int* in_sizes, int n_in,
                              void* d_out, int out_size, void* d_ws, size_t ws_size,
                              hipStream_t stream) {
    const float* x  = (const float*)d_in[0];
    const int*   ei = (const int*)d_in[1];
    const float* ew = (const float*)d_in[2];
    const float* w0 = (const float*)d_in[3];
    const float* b0 = (const float*)d_in[4];
    const float* w1 = (const float*)d_in[5];
    const float* b1 = (const float*)d_in[6];
    const float* w2 = (const float*)d_in[7];
    const float* b2 = (const float*)d_in[8];
    const float* w3 = (const float*)d_in[9];
    const float* b3 = (const float*)d_in[10];

    const int N = in_sizes[0] / 16;   // 50000
    const int E = in_sizes[2];        // 800000
    const int* src = ei;              // edge_index[0]
    const int* dst = ei + E;          // edge_index[1]

    float* dinv = (float*)d_ws;                  // N
    float* nrm  = dinv + N;                      // E
    float* bufA = nrm + E;                       // N*128
    float* bufB = bufA + (size_t)N * 128;        // N*128
    float* out  = (float*)d_out;                 // N*16

    const int TB = 256;
    const int gN = (N + TB - 1) / TB;
    const int gE = (E + TB - 1) / TB;

    // norm precompute (shared by all 4 layers)
    deg_init<<<gN, TB, 0, stream>>>(dinv, N);
    deg_edges<<<gE, TB, 0, stream>>>(dst, ew, dinv, E);
    make_dinv<<<gN, TB, 0, stream>>>(dinv, N);
    make_norm<<<gE, TB, 0, stream>>>(src, dst, ew, dinv, nrm, E);

    const int rowTiles   = (N + 15) / 16;
    const int gemmBlocks = (rowTiles + 7) / 8;
    const unsigned gInit128 = (unsigned)(((size_t)N * 128 + TB - 1) / TB);
    const unsigned gEdgeW   = (unsigned)((E + 7) / 8);   // wave per edge
    const unsigned gInit16  = (unsigned)(((size_t)N * 16 + TB - 1) / TB);
    const unsigned gEdge16  = (unsigned)(((size_t)E * 4 + TB - 1) / TB);

    // Layer 0: 16 -> 128
    gemm_wmma<16, 128, false><<<gemmBlocks, TB, 16 * 128 * sizeof(float), stream>>>(x, w0, bufA, N);
    agg_init<128><<<gInit128, TB, 0, stream>>>(bufA, dinv, b0, bufB, N);
    agg_edges128<<<gEdgeW, TB, 0, stream>>>(src, dst, nrm, bufA, bufB, E);

    // Layer 1: 128 -> 128 (relu folded into A-load)
    gemm_wmma<128, 128, true><<<gemmBlocks, TB, 128 * 128 * sizeof(float), stream>>>(bufB, w1, bufA, N);
    agg_init<128><<<gInit128, TB, 0, stream>>>(bufA, dinv, b1, bufB, N);
    agg_edges128<<<gEdgeW, TB, 0, stream>>>(src, dst, nrm, bufA, bufB, E);

    // Layer 2: 128 -> 128
    gemm_wmma<128, 128, true><<<gemmBlocks, TB, 128 * 128 * sizeof(float), stream>>>(bufB, w2, bufA, N);
    agg_init<128><<<gInit128, TB, 0, stream>>>(bufA, dinv, b2, bufB, N);
    agg_edges128<<<gEdgeW, TB, 0, stream>>>(src, dst, nrm, bufA, bufB, E);

    // Layer 3: 128 -> 16, aggregate straight into d_out
    gemm_wmma<128, 16, true><<<gemmBlocks, TB, 128 * 16 * sizeof(float), stream>>>(bufB, w3, bufA, N);
    agg_init<16><<<gInit16, TB, 0, stream>>>(bufA, dinv, b3, out, N);
    agg_edges<16><<<gEdge16, TB, 0, stream>>>(src, dst, nrm, bufA, out, E);
}